// Derivative1D_57492432224303
// MI455X (gfx1250) — compile-verified
//
#include <hip/hip_runtime.h>

// ---------------------------------------------------------------------------
// Derivative1D: y[b,i,:] = x[b,i+1,:] - x[b,i,:]   (B=64, L=16384, C=32, fp32)
// Pure-bandwidth kernel built on the CDNA5 async global->LDS path (ASYNCcnt).
// ---------------------------------------------------------------------------

#define B_   64
#define L_   16384
#define C_   32
#define LOUT (L_ - 1)                 // 16383 output rows per batch
#define ROW_BYTES (C_ * 4)            // 128 B per (b,l) row

#define WAVES_PER_BLOCK 8
#define ROWS_PER_WAVE   128
#define ROWS_PER_BLOCK  (WAVES_PER_BLOCK * ROWS_PER_WAVE)   // 1024
#define BLOCKS_PER_BATCH 16           // 16*1024 = 16384 >= 16383

#define CHUNK_ROWS 8                  // rows per pipeline stage
#define NCHUNK (ROWS_PER_WAVE / CHUNK_ROWS)  // 16 stages per wave
#define NBUF 4                        // LDS ring depth (per wave)
#define BUF_BYTES (CHUNK_ROWS * ROW_BYTES)   // 1024 B

#define LDS_AS __attribute__((address_space(3)))
#define GLB_AS __attribute__((address_space(1)))

typedef float v4f __attribute__((ext_vector_type(4)));
typedef int   v4i __attribute__((ext_vector_type(4)));

// ---- async global->LDS 128-bit load (CDNA5, ASYNCcnt-tracked) -------------
__device__ __forceinline__ void async_ld_b128(const void* gsrc, void* ldst) {
#if __has_builtin(__builtin_amdgcn_global_load_async_to_lds_b128)
  __builtin_amdgcn_global_load_async_to_lds_b128(
      (GLB_AS v4i*)gsrc, (LDS_AS v4i*)ldst, /*offset=*/0, /*cpol=*/0);
#else
  unsigned lds32 = (unsigned)(unsigned long long)(LDS_AS void*)ldst;
  asm volatile("global_load_async_to_lds_b128 %0, %1, off"
               :: "v"(lds32), "v"(gsrc) : "memory");
#endif
}

// ---- s_wait_asynccnt <= n  (compiler fence included) ----------------------
#if __has_builtin(__builtin_amdgcn_s_wait_asynccnt)
#define WAIT_ASYNC(n)                         \
  do {                                        \
    __builtin_amdgcn_s_wait_asynccnt(n);      \
    asm volatile("" ::: "memory");            \
  } while (0)
#else
#define WAIT_ASYNC(n)                                             \
  do {                                                            \
    asm volatile("s_wait_asynccnt %0" ::"n"(n) : "memory");       \
  } while (0)
#endif

#define WAIT_DS0() asm volatile("s_wait_dscnt 0x0" ::: "memory")

__global__ __launch_bounds__(256)
void Derivative1D_57492432224303_kernel(const float* __restrict__ x,
                                        float* __restrict__ y) {
  // Per-wave private ring buffers: 8 waves * 4 bufs * 1KB = 32 KB LDS.
  __shared__ __align__(16) unsigned char lds_raw[WAVES_PER_BLOCK * NBUF * BUF_BYTES];

  const int tid  = (int)threadIdx.x;
  const int wave = tid >> 5;
  const int lane = tid & 31;
  const int b    = (int)blockIdx.y;

  const int w0 = (int)blockIdx.x * ROWS_PER_BLOCK + wave * ROWS_PER_WAVE;
  const int w1 = min(w0 + ROWS_PER_WAVE, LOUT);   // only last wave is short by 1

  const unsigned char* xb = (const unsigned char*)(x) + (size_t)b * L_ * ROW_BYTES;
  unsigned char*       yb = (unsigned char*)(y) + (size_t)b * LOUT * ROW_BYTES;

  const unsigned wave_lds = (unsigned)(wave * NBUF * BUF_BYTES);

  // lane geometry for loads: 32 lanes * 16B = 4 rows per async instruction
  const int l_row = lane >> 3;          // row within a 4-row half-chunk
  const int l_col = (lane & 7) * 16;    // byte column within a 128B row

  // issue async loads for ring buffer k (input rows w0+8k .. w0+8k+7, clamped)
  auto issue_buf = [&](int k) {
    const int ib = w0 + k * CHUNK_ROWS;
    const unsigned lbase = wave_lds + (unsigned)((k & (NBUF - 1)) * BUF_BYTES);
#pragma unroll
    for (int t = 0; t < 2; ++t) {
      int row = ib + t * 4 + l_row;
      row = min(row, L_ - 1);           // clamp: keeps EXEC uniform, data unused
      const void* g = (const void*)(xb + (size_t)row * ROW_BYTES + l_col);
      void*       l = (void*)(lds_raw + lbase + (unsigned)(t * 512 + lane * 16));
      async_ld_b128(g, l);
    }
  };

  const v4f* lf4      = (const v4f*)lds_raw;
  const unsigned wb4  = wave_lds / 16;              // wave LDS base in v4f units
  const unsigned buf4 = BUF_BYTES / 16;             // 64 v4f per buffer

  // compute chunk i: outputs rows w0+8i .. +7 from buffers i and i+1
  auto compute = [&](int i) {
    const unsigned bi = (unsigned)(i & 3) * buf4;
    const unsigned bn = (unsigned)((i + 1) & 3) * buf4;
#pragma unroll
    for (int h = 0; h < 2; ++h) {
      const int p  = lane + h * 32;     // 0..63 v4f slots in the chunk
      const int j  = p >> 3;            // row within chunk (0..7)
      const int c4 = p & 7;             // v4f column
      v4f cur = lf4[wb4 + bi + (unsigned)(j * 8 + c4)];
      v4f nxt = (j < 7) ? lf4[wb4 + bi + (unsigned)((j + 1) * 8 + c4)]
                        : lf4[wb4 + bn + (unsigned)c4];
      const int r = w0 + i * CHUNK_ROWS + j;
      if (r < w1) {
        v4f d = nxt - cur;
        __builtin_nontemporal_store(
            d, (v4f*)(yb + (size_t)r * ROW_BYTES + (size_t)c4 * 16));
      }
    }
  };

  // prologue: fill 3 of 4 ring slots
  issue_buf(0);
  issue_buf(1);
  issue_buf(2);

  // steady state: issue buffer i+3, wait (<=4 outstanding => bufs 0..i+1 done)
#pragma unroll 2
  for (int i = 0; i < NCHUNK - 2; ++i) {     // i = 0..13, issues bufs 3..16
    WAIT_DS0();                              // ring slot reuse: prior LDS reads done
    issue_buf(i + 3);
    WAIT_ASYNC(4);
    compute(i);
  }

  // epilogue: everything resident
  WAIT_ASYNC(0);
  compute(NCHUNK - 2);
  compute(NCHUNK - 1);
}

extern "C" void kernel_launch(void* const* d_in, const int* in_sizes, int n_in,
                              void* d_out, int out_size, void* d_ws, size_t ws_size,
                              hipStream_t stream) {
  const float* x = (const float*)d_in[0];
  float*       y = (float*)d_out;
  (void)in_sizes; (void)n_in; (void)out_size; (void)d_ws; (void)ws_size;

  dim3 grid(BLOCKS_PER_BATCH, B_, 1);   // (16, 64)
  dim3 block(32 * WAVES_PER_BLOCK, 1, 1);
  Derivative1D_57492432224303_kernel<<<grid, block, 0, stream>>>(x, y);
}